// CAVAModule_79989470921163
// MI455X (gfx1250) — compile-verified
//
#include <hip/hip_runtime.h>
#include <hip/hip_bf16.h>

// ---------------------------------------------------------------------------
// Problem constants (match reference)
// ---------------------------------------------------------------------------
constexpr int kB    = 8;
constexpr int kT    = 2048;
constexpr int kVDIM = 1024;
constexpr int kADIM = 768;
constexpr int kDM   = 256;
constexpr int kHID  = 1024;
constexpr int kR    = kB * kT;          // 16384 rows total
constexpr float kEPS    = 1e-8f;
constexpr float kLNEPS  = 1e-5f;
constexpr float kWINDOW = 5.0f;

typedef __attribute__((ext_vector_type(2))) float v2f;
typedef __attribute__((ext_vector_type(8))) float v8f;
typedef int v4i_ __attribute__((vector_size(16)));   // matches builtin param pointee

// ---------------------------------------------------------------------------
// WMMA wrapper: D(16x16,f32) = A(16x4,f32) * B(4x16,f32) + C   (probe-verified)
// A frag: lane L (M = L&15) holds K = (L>=16 ? 2 : 0) in .x and +1 in .y
// B frag: lane L (N = L&15) holds K = (L>=16 ? 2 : 0) in .x and +1 in .y
// C/D:    VGPR i: lanes 0-15 -> M=i, lanes 16-31 -> M=i+8; N = lane&15
// ---------------------------------------------------------------------------
__device__ __forceinline__ v8f wmma_f32_k4(v2f a, v2f b, v8f c) {
  return __builtin_amdgcn_wmma_f32_16x16x4_f32(
      /*neg_a=*/false, a, /*neg_b=*/false, b,
      /*c_mod=*/(short)0, c, /*reuse_a=*/false, /*reuse_b=*/false);
}

__device__ __forceinline__ float wave_reduce_sum(float v) {
  v += __shfl_xor(v, 16, 32);
  v += __shfl_xor(v, 8, 32);
  v += __shfl_xor(v, 4, 32);
  v += __shfl_xor(v, 2, 32);
  v += __shfl_xor(v, 1, 32);
  return v;
}

// ---------------------------------------------------------------------------
// CDNA5 async global->LDS copy (ASYNCcnt-tracked), 16B per lane.
// Builtin signature (from hipcc diagnostic): arg0 = AS(1) int4*, and the LDS
// side is AS(3); AS(3) pointers are 32-bit and the LDS byte offset is the low
// 32 bits of the flat pointer (LDS aperture rule).
// ---------------------------------------------------------------------------
__device__ __forceinline__ void async_copy_b128(const float* g, float* l) {
#if __has_builtin(__builtin_amdgcn_global_load_async_to_lds_b128)
  __attribute__((address_space(1))) v4i_* gp =
      (__attribute__((address_space(1))) v4i_*)(unsigned long long)(uintptr_t)g;
  __attribute__((address_space(3))) v4i_* lp =
      (__attribute__((address_space(3))) v4i_*)(unsigned)(uintptr_t)l;
  __builtin_amdgcn_global_load_async_to_lds_b128(gp, lp, 0, 0);
#else
  asm volatile("global_load_async_to_lds_b128 %0, %1, off"
               :: "v"((unsigned)(uintptr_t)l), "v"(g)
               : "memory");
#endif
}

__device__ __forceinline__ void wait_async0() {
#if __has_builtin(__builtin_amdgcn_s_wait_asynccnt)
  __builtin_amdgcn_s_wait_asynccnt(0);
#else
  asm volatile("s_wait_asynccnt 0" ::: "memory");
#endif
}

// ---------------------------------------------------------------------------
// Kernel 1: out[r,:] = LayerNorm(in[r,:] @ W + bias), out is R x 256.
// One block = 32 rows (two M-halves share every B fragment -> 4 WMMA per
// 4 B-loads). 256 threads = 8 waves; each wave owns 2 col tiles x 2 M-halves.
// Input K-panels (32 x 512) staged in LDS via async b128 copies.
// Dynamic LDS: 32*512 + 32*256 floats = 96 KB (of 320 KB/WGP).
// ---------------------------------------------------------------------------
__global__ void proj_ln_kernel(const float* __restrict__ in,
                               const float* __restrict__ W,
                               const float* __restrict__ bias,
                               float* __restrict__ out, int K) {
  extern __shared__ float smem[];
  float* s_in  = smem;               // 32 * 512 panel
  float* s_out = smem + 32 * 512;    // 32 * 256 accum tile

  const int tid  = threadIdx.x;
  const int lane = tid & 31;
  const int wave = tid >> 5;
  const long r0  = (long)blockIdx.x * 32;

  const int m   = lane & 15;          // A-matrix row within a 16-row half
  const int kof = (lane >> 4) << 1;   // 0 for lanes 0-15, 2 for lanes 16-31
  const int n0  = (wave * 2) * 16 + (lane & 15);

  v8f acc00 = {}, acc01 = {};         // rows 0-15, col tiles 0/1
  v8f acc10 = {}, acc11 = {};         // rows 16-31, col tiles 0/1

  for (int kp = 0; kp < K; kp += 512) {
    const int klen = min(512, K - kp);
    // async-stage 32 x klen input panel, 16B granules, coalesced per row
    for (int idx = tid; idx < (32 * klen) / 4; idx += 256) {
      const int e   = idx * 4;
      const int row = e / klen;
      const int col = e - row * klen;
      async_copy_b128(&in[(r0 + row) * (long)K + kp + col],
                      &s_in[row * 512 + col]);
    }
    wait_async0();
    __syncthreads();

    for (int kk = 0; kk < klen; kk += 4) {
      v2f a0, a1;
      a0.x = s_in[m * 512 + kk + kof];
      a0.y = s_in[m * 512 + kk + kof + 1];
      a1.x = s_in[(16 + m) * 512 + kk + kof];
      a1.y = s_in[(16 + m) * 512 + kk + kof + 1];
      const float* Wp = W + (long)(kp + kk + kof) * kDM;
      __builtin_prefetch(Wp + 8 * kDM + n0, 0, 3);
      v2f b0; b0.x = Wp[n0];       b0.y = Wp[kDM + n0];
      v2f b1; b1.x = Wp[n0 + 16];  b1.y = Wp[kDM + n0 + 16];
      acc00 = wmma_f32_k4(a0, b0, acc00);
      acc10 = wmma_f32_k4(a1, b0, acc10);
      acc01 = wmma_f32_k4(a0, b1, acc01);
      acc11 = wmma_f32_k4(a1, b1, acc11);
    }
    __syncthreads();   // protect s_in before next panel load
  }

  // scatter accumulators (+bias) into the 32x256 LDS tile
  const int rbase = (lane >> 4) << 3;   // 0 or 8
  const float bA = bias[n0];
  const float bB = bias[n0 + 16];
#pragma unroll
  for (int i = 0; i < 8; ++i) {
    const int rm = i + rbase;
    s_out[rm * kDM + n0]             = acc00[i] + bA;
    s_out[rm * kDM + n0 + 16]        = acc01[i] + bB;
    s_out[(16 + rm) * kDM + n0]      = acc10[i] + bA;
    s_out[(16 + rm) * kDM + n0 + 16] = acc11[i] + bB;
  }
  __syncthreads();

  // LayerNorm over 256 columns: each wave normalizes 4 rows
#pragma unroll
  for (int rr = 0; rr < 4; ++rr) {
    const int row = wave * 4 + rr;
    float s = 0.f, s2 = 0.f;
#pragma unroll
    for (int j = 0; j < 8; ++j) {
      const float e = s_out[row * kDM + lane + j * 32];
      s += e;
      s2 += e * e;
    }
    s  = wave_reduce_sum(s);
    s2 = wave_reduce_sum(s2);
    const float mu  = s * (1.f / 256.f);
    const float var = s2 * (1.f / 256.f) - mu * mu;
    const float inv = rsqrtf(var + kLNEPS);
#pragma unroll
    for (int j = 0; j < 8; ++j) {
      const int c = lane + j * 32;
      out[(r0 + row) * (long)kDM + c] = (s_out[row * kDM + c] - mu) * inv;
    }
  }
}

// ---------------------------------------------------------------------------
// Kernel 2: fractional time-shift + causal window average.
// One block per (b,t); thread d handles feature channel d. The a-buffer
// (16.8 MB) is L2-resident on MI455X (192 MB L2).
// ---------------------------------------------------------------------------
__global__ void shift_ctx_kernel(const float* __restrict__ a,
                                 const float* __restrict__ theta,
                                 float* __restrict__ actx) {
  const int d  = threadIdx.x;      // 0..255
  const int bt = blockIdx.x;       // 0..B*T-1
  const int b  = bt / kT;
  const int t  = bt - b * kT;

  const float th    = fminf(fmaxf(theta[0], -12.f), 12.f);
  const float delta = 2.0f + 4.0f / (1.0f + expf(-th));
  const float dl    = fminf(fmaxf(delta, 0.f), (float)(kT - 1));
  const float nf    = floorf(dl);
  const float alpha = dl - nf;
  const int   ni    = (int)nf;

  const float center = fminf(fmaxf((float)t + delta, 0.f), (float)t);
  const int lo = max(0, (int)ceilf(center - kWINDOW));
  const int hi = min(t, (int)floorf(center + kWINDOW));

  const float* ab = a + (long)b * kT * kDM;
  float acc = 0.f;
  int cnt = 0;
  for (int tau = lo; tau <= hi; ++tau) {
    const int i0 = min(max(tau - ni, 0), kT - 1);
    const int i1 = min(i0 + 1, kT - 1);
    acc += (1.f - alpha) * ab[(long)i0 * kDM + d] + alpha * ab[(long)i1 * kDM + d];
    ++cnt;
  }
  actx[(long)bt * kDM + d] = acc / fmaxf((float)cnt, kEPS);
}

// ---------------------------------------------------------------------------
// x-fragment synthesis for the MLP GEMM:
// x = [an, vn, an*vn], built on the fly from raw v/a_ctx LDS tiles and
// per-row reciprocal L2 norms.
// ---------------------------------------------------------------------------
__device__ __forceinline__ v2f x_frag(const float* s_v, const float* s_a,
                                      const float* s_rv, const float* s_ra,
                                      int row, int k, int kof) {
  v2f a;
  if (k < kDM) {
    const int c = k + kof;
    a.x = s_a[row * kDM + c]     * s_ra[row];
    a.y = s_a[row * kDM + c + 1] * s_ra[row];
  } else if (k < 2 * kDM) {
    const int c = k - kDM + kof;
    a.x = s_v[row * kDM + c]     * s_rv[row];
    a.y = s_v[row * kDM + c + 1] * s_rv[row];
  } else {
    const int c = k - 2 * kDM + kof;
    const float sc = s_ra[row] * s_rv[row];
    a.x = s_a[row * kDM + c]     * s_v[row * kDM + c]     * sc;
    a.y = s_a[row * kDM + c + 1] * s_v[row * kDM + c + 1] * sc;
  }
  return a;
}

// ---------------------------------------------------------------------------
// Kernel 3: fused  x=[an,vn,an*vn]; h=gelu(x@W1+b1); g=clip(sig(clip(h@W2+b2)));
//           out = g*a_ctx + (1-g)*v.
// One block = 32 rows, 256 threads = 8 waves; each wave owns 8 col tiles of h
// across 2 M-halves (16 accumulators) -> 16 WMMA per 16 B-loads per k-step.
// Dynamic LDS: 2*32*256 + 96 floats ~= 64.4 KB.
// ---------------------------------------------------------------------------
__global__ void mlp_gate_kernel(const float* __restrict__ v,
                                const float* __restrict__ actx,
                                const float* __restrict__ W1,
                                const float* __restrict__ b1,
                                const float* __restrict__ W2,
                                const float* __restrict__ b2,
                                float* __restrict__ out) {
  extern __shared__ float smem[];
  float* s_v  = smem;               // 32 x 256 raw v tile
  float* s_a  = smem + 8192;        // 32 x 256 raw a_ctx tile
  float* s_rv = smem + 16384;       // 32 reciprocal L2 norms of v
  float* s_ra = s_rv + 32;          // 32 reciprocal L2 norms of a_ctx
  float* s_g  = s_ra + 32;          // 32 logits -> gates

  const int tid  = threadIdx.x;
  const int lane = tid & 31;
  const int wave = tid >> 5;
  const long r0  = (long)blockIdx.x * 32;

  // async-stage both 32x256 tiles (16B granules, coalesced)
  for (int idx = tid; idx < 2048; idx += 256) {
    const int e = idx * 4;
    async_copy_b128(&v[r0 * kDM + e],    &s_v[e]);
    async_copy_b128(&actx[r0 * kDM + e], &s_a[e]);
  }
  if (tid < 32) s_g[tid] = 0.f;
  wait_async0();
  __syncthreads();

  // reciprocal L2 norms: 8 waves x 4 rows
#pragma unroll
  for (int rr = 0; rr < 4; ++rr) {
    const int row = wave * 4 + rr;
    float sv = 0.f, sa = 0.f;
#pragma unroll
    for (int j = 0; j < 8; ++j) {
      const float ev = s_v[row * kDM + lane + j * 32];
      const float ea = s_a[row * kDM + lane + j * 32];
      sv += ev * ev;
      sa += ea * ea;
    }
    sv = wave_reduce_sum(sv);
    sa = wave_reduce_sum(sa);
    if (lane == 0) {
      s_rv[row] = 1.f / fmaxf(sqrtf(sv), kEPS);
      s_ra[row] = 1.f / fmaxf(sqrtf(sa), kEPS);
    }
  }
  __syncthreads();

  // GEMM: x(32x768) @ W1(768x1024)
  const int m   = lane & 15;
  const int kof = (lane >> 4) << 1;
  const int nl  = lane & 15;

  v8f accLo[8], accHi[8];
#pragma unroll
  for (int j = 0; j < 8; ++j) { accLo[j] = v8f{}; accHi[j] = v8f{}; }

  for (int k0 = 0; k0 < 3 * kDM; k0 += 4) {
    const v2f a0 = x_frag(s_v, s_a, s_rv, s_ra, m,      k0, kof);
    const v2f a1 = x_frag(s_v, s_a, s_rv, s_ra, 16 + m, k0, kof);
    const float* Wp = W1 + (long)(k0 + kof) * kHID;
    __builtin_prefetch(Wp + 8 * kHID + wave * 128 + nl, 0, 3);
#pragma unroll
    for (int j = 0; j < 8; ++j) {
      const int n = (wave * 8 + j) * 16 + nl;
      v2f bb; bb.x = Wp[n]; bb.y = Wp[kHID + n];
      accLo[j] = wmma_f32_k4(a0, bb, accLo[j]);
      accHi[j] = wmma_f32_k4(a1, bb, accHi[j]);
    }
  }

  // exact GELU + fold h @ W2 via half-wave shuffle reduce + LDS float atomics
  const int rbase = (lane >> 4) << 3;   // C/D rows 0-7 or 8-15 of each half
  const float inv_sqrt2 = 0.70710678118654752440f;
#pragma unroll
  for (int j = 0; j < 8; ++j) {
    const int n = (wave * 8 + j) * 16 + nl;
    const float w2  = W2[n];
    const float bb1 = b1[n];
#pragma unroll
    for (int i = 0; i < 8; ++i) {
      const float hLo = accLo[j][i] + bb1;
      const float hHi = accHi[j][i] + bb1;
      float p0 = 0.5f * hLo * (1.0f + erff(hLo * inv_sqrt2)) * w2;
      float p1 = 0.5f * hHi * (1.0f + erff(hHi * inv_sqrt2)) * w2;
      p0 += __shfl_xor(p0, 1, 32);  p1 += __shfl_xor(p1, 1, 32);
      p0 += __shfl_xor(p0, 2, 32);  p1 += __shfl_xor(p1, 2, 32);
      p0 += __shfl_xor(p0, 4, 32);  p1 += __shfl_xor(p1, 4, 32);
      p0 += __shfl_xor(p0, 8, 32);  p1 += __shfl_xor(p1, 8, 32);
      if ((lane & 15) == 0) {
        atomicAdd(&s_g[i + rbase],      p0);   // ds_add_f32
        atomicAdd(&s_g[16 + i + rbase], p1);
      }
    }
  }
  __syncthreads();

  if (tid < 32) {
    const float logit = fminf(fmaxf(s_g[tid] + b2[0], -12.f), 12.f);
    const float g = 1.0f / (1.0f + expf(-logit));
    s_g[tid] = fminf(fmaxf(g, 0.05f), 0.95f);
  }
  __syncthreads();

  for (int idx = tid; idx < 8192; idx += 256) {
    const int row = idx >> 8;
    const float g = s_g[row];
    out[r0 * kDM + idx] = g * s_a[idx] + (1.f - g) * s_v[idx];
  }
}

// ---------------------------------------------------------------------------
// Host launcher
// ---------------------------------------------------------------------------
extern "C" void kernel_launch(void* const* d_in, const int* in_sizes, int n_in,
                              void* d_out, int out_size, void* d_ws, size_t ws_size,
                              hipStream_t stream) {
  const float* video = (const float*)d_in[0];
  const float* audio = (const float*)d_in[1];
  const float* Wv    = (const float*)d_in[2];
  const float* bv    = (const float*)d_in[3];
  const float* Wa    = (const float*)d_in[4];
  const float* ba    = (const float*)d_in[5];
  const float* theta = (const float*)d_in[6];
  const float* W1    = (const float*)d_in[7];
  const float* b1    = (const float*)d_in[8];
  const float* W2    = (const float*)d_in[9];
  const float* b2    = (const float*)d_in[10];
  float* out = (float*)d_out;

  float* v_buf = (float*)d_ws;                       // R x 256
  float* a_buf = v_buf + (size_t)kR * kDM;           // R x 256
  float* actx  = a_buf + (size_t)kR * kDM;           // R x 256

  const size_t lds_proj = (size_t)(32 * 512 + 32 * kDM) * sizeof(float);   // 96 KB
  const size_t lds_mlp  = (size_t)(2 * 32 * kDM + 96) * sizeof(float);     // ~64.4 KB

  proj_ln_kernel<<<kR / 32, 256, lds_proj, stream>>>(video, Wv, bv, v_buf, kVDIM);
  proj_ln_kernel<<<kR / 32, 256, lds_proj, stream>>>(audio, Wa, ba, a_buf, kADIM);
  shift_ctx_kernel<<<kR, 256, 0, stream>>>(a_buf, theta, actx);
  mlp_gate_kernel<<<kR / 32, 256, lds_mlp, stream>>>(v_buf, actx, W1, b1, W2, b2, out);
}